// PCLLoss_10058813407513
// MI455X (gfx1250) — compile-verified
//
#include <hip/hip_runtime.h>

typedef __attribute__((ext_vector_type(2))) float v2f;
typedef __attribute__((ext_vector_type(8))) float v8f;

// ---------------------------------------------------------------------------
// Wave-level sum using V_WMMA_F32_16X16X4_F32 (exact fp32, RNE).
// A (16x4 f32) = 2 VGPRs/lane: slot0 = this lane's partial, slot1 = 0.
// B (4x16)    = all ones. C = 0.  D[m][n] = row-sum of A.
// Per the 16x16 f32 C/D layout: VGPR r holds M=r (lanes 0-15) / M=8+r
// (lanes 16-31), so summing a lane's 8 D VGPRs gives sum over 8 distinct M;
// lane ^ 16 holds the other 8 rows -> one shfl_xor finishes the wave total.
// Requires EXEC == all 1s (call only from fully converged code).
// ---------------------------------------------------------------------------
__device__ __forceinline__ float wave_sum_wmma(float s) {
  v2f a; a.x = s;    a.y = 0.0f;
  v2f b; b.x = 1.0f; b.y = 1.0f;
  v8f c = {};
  v8f d = __builtin_amdgcn_wmma_f32_16x16x4_f32(
      /*neg_a=*/false, a, /*neg_b=*/false, b,
      /*c_mod=*/(short)0, c, /*reuse_a=*/false, /*reuse_b=*/false);
  float t = 0.0f;
#pragma unroll
  for (int i = 0; i < 8; ++i) t += d[i];
  t += __shfl_xor(t, 16, 32);   // wave32: pair lanes L and L^16
  return t;                     // every lane holds the wave total
}

// Block-level sum (256 threads = 8 waves on wave32). Result valid on thread 0.
__device__ __forceinline__ float block_sum(float s) {
  __shared__ float lds[8];
  float w = wave_sum_wmma(s);
  const int lane = threadIdx.x & 31;
  const int wid  = threadIdx.x >> 5;
  if (lane == 0) lds[wid] = w;
  __syncthreads();
  float total = 0.0f;
  if (threadIdx.x == 0) {
    const int nw = (blockDim.x + 31) >> 5;
    for (int i = 0; i < nw; ++i) total += lds[i];   // fixed order: deterministic
  }
  return total;
}

// ---------------------------------------------------------------------------
// bg term: sum over ROIs with labels==0 (and im_labels[0]!=0) of
//          w[i] * log(pcl_probs[i, 0]).  Column-0 gather, stride C floats.
// ---------------------------------------------------------------------------
__global__ void __launch_bounds__(256)
bg_partials_kernel(const float* __restrict__ probs,
                   const int*   __restrict__ labels,
                   const float* __restrict__ clw,
                   const int*   __restrict__ im_labels,
                   float* __restrict__ partials,
                   int n, int stride_c) {
  const int  im0 = im_labels[0];
  long long  i   = (long long)blockIdx.x * blockDim.x + threadIdx.x;
  const long long gs = (long long)gridDim.x * blockDim.x;
  float s = 0.0f;
  for (; i < n; i += gs) {
    // hide latency of the strided (84 B) column gather: prefetch next stride
    if (i + gs < n)
      __builtin_prefetch(probs + (i + gs) * stride_c, 0, 0);
    const int   lbl = labels[i];
    const float w   = clw[i];
    const float p   = probs[i * stride_c];
    if (lbl == 0 && im0 != 0) s += w * logf(p);
  }
  const float t = block_sum(s);
  if (threadIdx.x == 0) partials[blockIdx.x] = t;
}

// ---------------------------------------------------------------------------
// fg term: sum over clusters k with im_labels[pc_labels[k]] != 0 of
//          img_w[k] * log(pc_probs[k]).  K = 4096 -> one block.
// ---------------------------------------------------------------------------
__global__ void __launch_bounds__(256)
fg_partials_kernel(const float* __restrict__ pc_probs,
                   const int*   __restrict__ pc_labels,
                   const float* __restrict__ img_w,
                   const int*   __restrict__ im_labels,
                   float* __restrict__ partials,
                   int k, int slot) {
  float s = 0.0f;
  for (int i = threadIdx.x; i < k; i += blockDim.x) {
    const int cl = pc_labels[i];
    if (im_labels[cl] != 0) s += img_w[i] * logf(pc_probs[i]);
  }
  const float t = block_sum(s);
  if (threadIdx.x == 0) partials[slot] = t;
}

// ---------------------------------------------------------------------------
// Final: out = -(sum of partials) / N, fixed-order -> deterministic.
// ---------------------------------------------------------------------------
__global__ void __launch_bounds__(256)
finalize_kernel(const float* __restrict__ partials, int m,
                float neg_inv_n, float* __restrict__ out) {
  float s = 0.0f;
  for (int i = threadIdx.x; i < m; i += blockDim.x) s += partials[i];
  const float t = block_sum(s);
  if (threadIdx.x == 0) out[0] = t * neg_inv_n;
}

extern "C" void kernel_launch(void* const* d_in, const int* in_sizes, int n_in,
                              void* d_out, int out_size, void* d_ws, size_t ws_size,
                              hipStream_t stream) {
  // setup_inputs() order:
  // 0 pcl_probs (N*C f32), 1 labels (N i32), 2 cls_loss_weights (N f32),
  // 3 gt_assignment (unused), 4 pc_labels (K i32), 5 pc_probs (K f32),
  // 6 pc_count (unused), 7 img_cls_loss_weights (K f32), 8 im_labels (C i32)
  const float* probs     = (const float*)d_in[0];
  const int*   labels    = (const int*)  d_in[1];
  const float* clw       = (const float*)d_in[2];
  const int*   pc_labels = (const int*)  d_in[4];
  const float* pc_probs  = (const float*)d_in[5];
  const float* img_w     = (const float*)d_in[7];
  const int*   im_labels = (const int*)  d_in[8];

  const int n = in_sizes[1];   // N
  const int c = in_sizes[8];   // C (row stride of pcl_probs)
  const int k = in_sizes[4];   // K

  int NB = 1024;               // bg partial blocks
  if (ws_size < (size_t)(NB + 1) * sizeof(float)) {
    NB = (int)(ws_size / sizeof(float)) - 1;
    if (NB < 1) NB = 1;
  }
  float* partials = (float*)d_ws;

  bg_partials_kernel<<<NB, 256, 0, stream>>>(probs, labels, clw, im_labels,
                                             partials, n, c);
  fg_partials_kernel<<<1, 256, 0, stream>>>(pc_probs, pc_labels, img_w,
                                            im_labels, partials, k, NB);
  finalize_kernel<<<1, 256, 0, stream>>>(partials, NB + 1,
                                         -1.0f / (float)n, (float*)d_out);
}